// PSMIL_22239340659264
// MI455X (gfx1250) — compile-verified
//
#include <hip/hip_runtime.h>
#include <math.h>

typedef float v2f __attribute__((ext_vector_type(2)));
typedef float v8f __attribute__((ext_vector_type(8)));

#define L_DIM 512
#define RPB1  128   // rows per block, pass1 (8 waves * 16 rows)
#define KC    32    // K chunk staged in LDS per iteration
#define XPAD  36    // padded row stride (words) for x tile: 36*r mod 64 -> conflict-free
#define WPAD  516   // padded row stride (words) for weight block: 4-apart banks
#define RPB3  256   // rows per block, pass2

// CDNA5 async global->LDS copy (ASYNCcnt-tracked, no VGPR round trip).
// Guarded so a missing builtin cleanly falls back to the sync copy path.
#if defined(__has_builtin)
#if __has_builtin(__builtin_amdgcn_global_load_async_to_lds_b128)
#define HAVE_ASYNC_LDS 1
#endif
#endif

#ifdef HAVE_ASYNC_LDS
// Builtin prototype (from hipcc diagnostic): (v4i AS1*, v4i AS3*, imm, imm)
typedef int v4i_vs __attribute__((vector_size(16)));
typedef __attribute__((address_space(1))) v4i_vs g_v4i;
typedef __attribute__((address_space(3))) v4i_vs l_v4i;

__device__ __forceinline__ void async_copy16(const float* g, float* l) {
    __builtin_amdgcn_global_load_async_to_lds_b128(
        (g_v4i*)(uintptr_t)g, (l_v4i*)(uintptr_t)l, 0, 0);
}
__device__ __forceinline__ void wait_async0() {
#if __has_builtin(__builtin_amdgcn_s_wait_asynccnt)
    __builtin_amdgcn_s_wait_asynccnt(0);
#else
    asm volatile("s_wait_asynccnt 0" ::: "memory");
#endif
}
#endif

// ---------------------------------------------------------------------------
// Pass 1: WMMA f32 16x16x4 GEMM of x (N x 512) against [linear | fbank | 0pad]
// (512 x 16). Produces ins_probs, attention scores s, per-block partial
// reductions (max s, argmax predy).
// ---------------------------------------------------------------------------
__global__ __launch_bounds__(256) void psmil_pass1(
    const float* __restrict__ x, const float* __restrict__ y,
    const float* __restrict__ linear, const float* __restrict__ fbank,
    const float* __restrict__ w_ps, const float* __restrict__ b_ps,
    int N,
    float* __restrict__ out_ins,
    float* __restrict__ ws_s,
    float* __restrict__ ws_smax,
    float* __restrict__ ws_pmaxv,
    int*   __restrict__ ws_pmaxi)
{
    __shared__ float lds_w[16 * WPAD];   // B operand: 16 cols (4 live) x 512 K
    __shared__ float lds_x[RPB1 * XPAD]; // A tile: 128 rows x KC cols (padded)
    __shared__ float lds_d[8 * 256];     // per-wave 16x16 D dump
    __shared__ float red_s[RPB1];
    __shared__ float red_pv[RPB1];
    __shared__ int   red_pi[RPB1];

    const int tid  = threadIdx.x;
    const int wave = tid >> 5;
    const int lane = tid & 31;
    const int half = lane >> 4;   // lanes 16-31 hold K+2,K+3 (A) / rows K+2,K+3 (B)
    const int l16  = lane & 15;
    const int rowBase = blockIdx.x * RPB1;
    const bool fullBlock = (rowBase + RPB1 <= N);

    // Stage weight block in B-operand-friendly layout: lds_w[n][k], zero-padded n>=4
    for (int i = tid; i < 16 * WPAD; i += 256) {
        int n = i / WPAD, k = i - n * WPAD;
        float v = 0.0f;
        if (k < L_DIM) {
            if (n < 2)      v = linear[k * 2 + n];
            else if (n < 4) v = fbank[k * 2 + (n - 2)];
        }
        lds_w[i] = v;
    }

    v8f acc = {0.f, 0.f, 0.f, 0.f, 0.f, 0.f, 0.f, 0.f};

    for (int k0 = 0; k0 < L_DIM; k0 += KC) {
        __syncthreads();   // previous tile fully consumed by all waves
        if (fullBlock) {
            // Hot path: no bounds checks; CDNA5 async copy when available.
            #pragma unroll
            for (int u = 0; u < 4; ++u) {
                const int t  = tid + u * 256;
                const int r  = t >> 3;          // KC/4 == 8 float4 per row
                const int c4 = (t & 7) * 4;
                const float* gp = x + (size_t)(rowBase + r) * L_DIM + k0 + c4;
                float*       lp = lds_x + r * XPAD + c4;
#ifdef HAVE_ASYNC_LDS
                async_copy16(gp, lp);
#else
                *(float4*)lp = *(const float4*)gp;
#endif
            }
#ifdef HAVE_ASYNC_LDS
            wait_async0();
#endif
        } else {
            // Tail block only: guarded, zero-filled.
            #pragma unroll
            for (int u = 0; u < 4; ++u) {
                const int t  = tid + u * 256;
                const int r  = t >> 3;
                const int c4 = (t & 7) * 4;
                const int row = rowBase + r;
                float4 v = make_float4(0.f, 0.f, 0.f, 0.f);
                if (row < N)
                    v = *(const float4*)(x + (size_t)row * L_DIM + k0 + c4);
                *(float4*)(lds_x + r * XPAD + c4) = v;
            }
        }
        __syncthreads();
        // A: lane l16 -> row (wave*16+l16); VGPR0/1 = K+0,K+1 (lanes 0-15), K+2,K+3 (16-31)
        const float* ax = lds_x + (wave * 16 + l16) * XPAD + half * 2;
        // B: lane l16 -> col n; rows K+0,K+1 (lanes 0-15), K+2,K+3 (lanes 16-31)
        const float* bw = lds_w + l16 * WPAD + k0 + half * 2;
        #pragma unroll
        for (int kk = 0; kk < KC; kk += 4) {
            v2f a = *(const v2f*)(ax + kk);
            v2f b = *(const v2f*)(bw + kk);
            acc = __builtin_amdgcn_wmma_f32_16x16x4_f32(
                false, a, false, b, (short)0, acc, false, false);
        }
    }

    // Dump D (16x16) to LDS: VGPR r, lanes 0-15 -> M=r, lanes 16-31 -> M=8+r, N=l16
    {
        float* dd = lds_d + wave * 256;
        const int m0 = half * 8;
        #pragma unroll
        for (int r = 0; r < 8; ++r)
            dd[(m0 + r) * 16 + l16] = acc[r];
    }
    __syncthreads();

    const int   yidx = (y[1] > y[0]) ? 1 : 0;
    const float w0 = w_ps[0], w1 = w_ps[1], bp = b_ps[0];
    const int   lr = wave * 16 + l16;   // 0..127
    if (half == 0) {
        const int row = rowBase + lr;
        const float* dd = lds_d + wave * 256 + l16 * 16;
        float d0 = dd[0], d1 = dd[1], e0 = dd[2], e1 = dd[3];
        // ins_probs = softmax([d0,d1])
        float md = fmaxf(d0, d1);
        float p0 = expf(d0 - md), p1 = expf(d1 - md);
        float invp = 1.0f / (p0 + p1);
        // pred = softmax([e0,e1]); score s = pred . w_ps + b_ps
        float me = fmaxf(e0, e1);
        float q0 = expf(e0 - me), q1 = expf(e1 - me);
        float invq = 1.0f / (q0 + q1);
        q0 *= invq; q1 *= invq;
        float s = q0 * w0 + q1 * w1 + bp;
        float predy = yidx ? q1 : q0;
        if (row < N) {
            out_ins[(size_t)row * 2 + 0] = p0 * invp;
            out_ins[(size_t)row * 2 + 1] = p1 * invp;
            ws_s[row] = s;
            red_s[lr] = s; red_pv[lr] = predy; red_pi[lr] = row;
        } else {
            red_s[lr] = -INFINITY; red_pv[lr] = -INFINITY; red_pi[lr] = 0x7FFFFFFF;
        }
    }
    __syncthreads();
    for (int st = 64; st > 0; st >>= 1) {
        if (tid < st) {
            red_s[tid] = fmaxf(red_s[tid], red_s[tid + st]);
            float pv = red_pv[tid + st]; int pi = red_pi[tid + st];
            if (pv > red_pv[tid] || (pv == red_pv[tid] && pi < red_pi[tid])) {
                red_pv[tid] = pv; red_pi[tid] = pi;
            }
        }
        __syncthreads();
    }
    if (tid == 0) {
        ws_smax[blockIdx.x]  = red_s[0];
        ws_pmaxv[blockIdx.x] = red_pv[0];
        ws_pmaxi[blockIdx.x] = red_pi[0];
    }
}

// ---------------------------------------------------------------------------
// Global reductions: Smax, sel = first argmax(predy), sumexp (fixed order).
// ---------------------------------------------------------------------------
__global__ __launch_bounds__(1024) void psmil_reduce(
    const float* __restrict__ ws_s,
    const float* __restrict__ ws_smax,
    const float* __restrict__ ws_pmaxv,
    const int*   __restrict__ ws_pmaxi,
    int NB1, int N, float* __restrict__ sc)
{
    __shared__ float rs[1024];
    __shared__ float rv[1024];
    __shared__ int   ri[1024];
    const int tid = threadIdx.x;

    float sm = -INFINITY, pv = -INFINITY; int pi = 0x7FFFFFFF;
    for (int i = tid; i < NB1; i += 1024) {
        sm = fmaxf(sm, ws_smax[i]);
        float v = ws_pmaxv[i]; int ix = ws_pmaxi[i];
        if (v > pv || (v == pv && ix < pi)) { pv = v; pi = ix; }
    }
    rs[tid] = sm; rv[tid] = pv; ri[tid] = pi;
    __syncthreads();
    for (int st = 512; st > 0; st >>= 1) {
        if (tid < st) {
            rs[tid] = fmaxf(rs[tid], rs[tid + st]);
            float v = rv[tid + st]; int ix = ri[tid + st];
            if (v > rv[tid] || (v == rv[tid] && ix < ri[tid])) { rv[tid] = v; ri[tid] = ix; }
        }
        __syncthreads();
    }
    const float Smax = rs[0];
    const int   sel  = ri[0];
    __syncthreads();

    float local = 0.0f;
    for (int i = tid; i < N; i += 1024) local += expf(ws_s[i] - Smax);
    rs[tid] = local;
    __syncthreads();
    for (int st = 512; st > 0; st >>= 1) {
        if (tid < st) rs[tid] += rs[tid + st];
        __syncthreads();
    }
    if (tid == 0) {
        sc[0] = Smax;
        sc[1] = rs[0];
        ((int*)sc)[2] = sel;
    }
}

// ---------------------------------------------------------------------------
// Pass 2: alpha out; per-block partial Fagg = sum(alpha_i * x_i) and colsum.
// Fully coalesced float2 reads of x (thread t owns columns 2t, 2t+1).
// ---------------------------------------------------------------------------
__global__ __launch_bounds__(256) void psmil_pass2(
    const float* __restrict__ x, const float* __restrict__ ws_s,
    const float* __restrict__ sc, int N,
    float* __restrict__ out_alpha,
    float* __restrict__ ws_fagg, float* __restrict__ ws_csum)
{
    __shared__ float la[RPB3];
    const int tid = threadIdx.x;
    const int blk = blockIdx.x;
    const float Smax = sc[0];
    const float invZ = 1.0f / sc[1];

    const int i = blk * RPB3 + tid;
    float a = 0.0f;
    if (i < N) {
        a = expf(ws_s[i] - Smax) * invZ;
        out_alpha[i] = a;
    }
    la[tid] = a;
    __syncthreads();

    const int j0 = tid * 2;
    float f0 = 0.f, f1 = 0.f, c0 = 0.f, c1 = 0.f;
    int rmax = N - blk * RPB3;
    if (rmax > RPB3) rmax = RPB3;
    for (int r = 0; r < rmax; ++r) {
        const float2 v = *(const float2*)(x + (size_t)(blk * RPB3 + r) * L_DIM + j0);
        const float ar = la[r];
        f0 += ar * v.x; f1 += ar * v.y;
        c0 += v.x;      c1 += v.y;
    }
    ws_fagg[(size_t)blk * L_DIM + j0]     = f0;
    ws_fagg[(size_t)blk * L_DIM + j0 + 1] = f1;
    ws_csum[(size_t)blk * L_DIM + j0]     = c0;
    ws_csum[(size_t)blk * L_DIM + j0 + 1] = c1;
}

// ---------------------------------------------------------------------------
// Final: Fagg, fbank EMA column update + normalize (+NaN fallback to column
// mean), Y_prob = log_softmax(Fagg @ linear), Y_hat.
// ---------------------------------------------------------------------------
__global__ __launch_bounds__(512) void psmil_final(
    const float* __restrict__ x, const float* __restrict__ y,
    const float* __restrict__ linear, const float* __restrict__ fbank,
    const float* __restrict__ sc, const float* __restrict__ ws_fagg,
    const float* __restrict__ ws_csum, int NB3, int N,
    float* __restrict__ out_Fagg, float* __restrict__ out_nfb,
    float* __restrict__ out_Yprob, float* __restrict__ out_Yhat)
{
    __shared__ float red[512];
    __shared__ int   redi[512];
    __shared__ float Lsh[1];
    const int j = threadIdx.x;

    float fa = 0.f, cs = 0.f;
    for (int b = 0; b < NB3; ++b) {           // fixed order -> deterministic
        fa += ws_fagg[(size_t)b * L_DIM + j];
        cs += ws_csum[(size_t)b * L_DIM + j];
    }
    out_Fagg[j] = fa;
    const float colmean = cs / (float)N;

    const int   yidx = (y[1] > y[0]) ? 1 : 0;
    const int   sel  = ((const int*)sc)[2];
    const float cf   = x[(size_t)sel * L_DIM + j];
    const float v    = 0.99f * fbank[j * 2 + yidx] + 0.01f * cf;

    red[j] = v * v;
    __syncthreads();
    for (int st = 256; st > 0; st >>= 1) { if (j < st) red[j] += red[j + st]; __syncthreads(); }
    const float nrm = sqrtf(red[0]);
    __syncthreads();
    float newfy = v / fmaxf(nrm, 1e-12f);

    redi[j] = isnan(newfy) ? 1 : 0;
    __syncthreads();
    for (int st = 256; st > 0; st >>= 1) { if (j < st) redi[j] |= redi[j + st]; __syncthreads(); }
    if (redi[0]) newfy = colmean;

    out_nfb[j * 2 + yidx]       = newfy;
    out_nfb[j * 2 + (1 - yidx)] = fbank[j * 2 + (1 - yidx)];

    red[j] = fa * linear[j * 2 + 0];
    __syncthreads();
    for (int st = 256; st > 0; st >>= 1) { if (j < st) red[j] += red[j + st]; __syncthreads(); }
    if (j == 0) Lsh[0] = red[0];
    __syncthreads();
    red[j] = fa * linear[j * 2 + 1];
    __syncthreads();
    for (int st = 256; st > 0; st >>= 1) { if (j < st) red[j] += red[j + st]; __syncthreads(); }
    if (j == 0) {
        const float L0 = Lsh[0], L1 = red[0];
        const float m   = fmaxf(L0, L1);
        const float lse = m + logf(expf(L0 - m) + expf(L1 - m));
        out_Yprob[0] = L0 - lse;
        out_Yprob[1] = L1 - lse;
        out_Yhat[0]  = (L1 > L0) ? 1.0f : 0.0f;
    }
}

// ---------------------------------------------------------------------------
extern "C" void kernel_launch(void* const* d_in, const int* in_sizes, int n_in,
                              void* d_out, int out_size, void* d_ws, size_t ws_size,
                              hipStream_t stream)
{
    const float* x      = (const float*)d_in[0];
    const float* y      = (const float*)d_in[1];
    const float* linear = (const float*)d_in[3];
    const float* fbank  = (const float*)d_in[4];
    const float* w_ps   = (const float*)d_in[5];
    const float* b_ps   = (const float*)d_in[6];

    const int N   = in_sizes[0] / L_DIM;
    const int NB1 = (N + RPB1 - 1) / RPB1;
    const int NB3 = (N + RPB3 - 1) / RPB3;

    // Output layout (tuple flattened in return order):
    // Y_prob(2) | Y_hat(1) | alpha(N) | ins_probs(2N) | Fagg(512) | new_fbank(1024)
    float* out        = (float*)d_out;
    float* out_Yprob  = out;
    float* out_Yhat   = out + 2;
    float* out_alpha  = out + 3;
    float* out_ins    = out + 3 + (size_t)N;
    float* out_Fagg   = out + 3 + 3 * (size_t)N;
    float* out_nfb    = out + 3 + 3 * (size_t)N + L_DIM;

    // Workspace layout (floats)
    float* ws       = (float*)d_ws;
    float* ws_s     = ws;                        // N
    float* ws_smax  = ws + N;                    // NB1
    float* ws_pmaxv = ws + N + NB1;              // NB1
    int*   ws_pmaxi = (int*)(ws + N + 2 * NB1);  // NB1
    float* sc       = ws + N + 3 * NB1;          // 8 scalars
    float* ws_fagg  = sc + 8;                    // NB3*512
    float* ws_csum  = ws_fagg + (size_t)NB3 * L_DIM; // NB3*512

    psmil_pass1<<<NB1, 256, 0, stream>>>(x, y, linear, fbank, w_ps, b_ps, N,
                                         out_ins, ws_s, ws_smax, ws_pmaxv, ws_pmaxi);
    psmil_reduce<<<1, 1024, 0, stream>>>(ws_s, ws_smax, ws_pmaxv, ws_pmaxi, NB1, N, sc);
    psmil_pass2<<<NB3, 256, 0, stream>>>(x, ws_s, sc, N, out_alpha, ws_fagg, ws_csum);
    psmil_final<<<1, 512, 0, stream>>>(x, y, linear, fbank, sc, ws_fagg, ws_csum,
                                       NB3, N, out_Fagg, out_nfb, out_Yprob, out_Yhat);
}